// VarianceAdaptor_80711025427129
// MI455X (gfx1250) — compile-verified
//
#include <hip/hip_runtime.h>
#include <hip/hip_bf16.h>

typedef __attribute__((ext_vector_type(2))) float v2f;
typedef __attribute__((ext_vector_type(8))) float v8f;

#define BM 64
#define BN 128
#define BK 32
#define APAD 4   // As row stride 36 floats: conflict-free, rows 16B-aligned
#define BPAD 8   // Bs row stride 136 floats: half-wave rows hit disjoint banks

#define ASYNC_B128(ldsoff, gaddr) \
    asm volatile("global_load_async_to_lds_b128 %0, %1, off" \
                 :: "v"(ldsoff), "v"(gaddr) : "memory")

// ---------------------------------------------------------------------------
// Tiled fp32 WMMA GEMM with double-buffered async global->LDS staging.
//   Y[M,N] (+)= X3[M,K] * Wt[K,N] + bias[N]
//   conv==1: X is PRE-PADDED (B, Tp=T+2, Cin); the k=3 im2col row for (b,t)
//            is the contiguous window Xpad[(b*Tp+t)*Cin .. +3*Cin) -> the
//            A-tile loads are contiguous async B128s, branch-free.
//   conv==0: X is plain M x K row-major (m clamped to M-1: branch-free).
//   opad==1: output rows stored in padded (B, T+2, N) layout (row m+2b+1).
// Block: 256 threads = 8 waves; block tile 64x128; wave tile 32x32.
// BK=32 => 32 v_wmma (8 k-steps x 2x2 tiles) per barrier pair per wave.
// Pipeline unrolled by 2 so double-buffer indices are compile-time constants.
// Requires K % 64 == 0 (all call sites: K = 768, 3072, 256).
// ---------------------------------------------------------------------------
__global__ __launch_bounds__(256)
void wmma_gemm(const float* __restrict__ X, const float* __restrict__ Wt,
               const float* __restrict__ bias, float* __restrict__ Y,
               int M, int T, int Tp, int cshift, int N, int K,
               int conv, int opad, int accum)
{
    __shared__ float As[2][BM][BK + APAD];
    __shared__ float Bs[2][BK][BN + BPAD];

    const int tid  = threadIdx.x;
    const int lane = tid & 31;
    const int wave = tid >> 5;
    const int wm   = (wave >> 2) * 32;   // 2 waves along M
    const int wn   = (wave & 3)  * 32;   // 4 waves along N
    const int m0   = blockIdx.y * BM;
    const int n0   = blockIdx.x * BN;

    // ---- per-thread async staging assignments (fixed for whole K loop) ----
    // A tile: 64 rows x 8 quads = 512 B128 loads, two per thread
    //         (quads kq and kq+4 of row mi).
    const int mi = tid >> 2;
    const int kq = tid & 3;
    int m_cl = m0 + mi; if (m_cl > M - 1) m_cl = M - 1;
    size_t arow;
    if (conv) {
        int b = m_cl / T;
        int t = m_cl - b * T;
        arow = ((size_t)(b * Tp + t)) << cshift;   // padded row base
    } else {
        arow = (size_t)m_cl * K;
    }
    const float* pA = X + arow + (kq << 2);
    const unsigned ldsA[2][2] = {
        { (unsigned)(uintptr_t)&As[0][mi][kq * 4],
          (unsigned)(uintptr_t)&As[0][mi][kq * 4 + 16] },
        { (unsigned)(uintptr_t)&As[1][mi][kq * 4],
          (unsigned)(uintptr_t)&As[1][mi][kq * 4 + 16] } };

    // B tile: 32 rows x 32 quads = 1024 B128 loads, four per thread
    //         (rows r0+8j, quad q0).
    const int r0 = tid >> 5;
    const int q0 = tid & 31;
    const float* pB = Wt + (size_t)r0 * N + n0 + q0 * 4;
    const size_t  b8N = (size_t)8 * N;
    const unsigned ldsB[2][4] = {
        { (unsigned)(uintptr_t)&Bs[0][r0][q0 * 4],
          (unsigned)(uintptr_t)&Bs[0][r0 +  8][q0 * 4],
          (unsigned)(uintptr_t)&Bs[0][r0 + 16][q0 * 4],
          (unsigned)(uintptr_t)&Bs[0][r0 + 24][q0 * 4] },
        { (unsigned)(uintptr_t)&Bs[1][r0][q0 * 4],
          (unsigned)(uintptr_t)&Bs[1][r0 +  8][q0 * 4],
          (unsigned)(uintptr_t)&Bs[1][r0 + 16][q0 * 4],
          (unsigned)(uintptr_t)&Bs[1][r0 + 24][q0 * 4] } };

    // buf is always a literal at the call sites -> branches fold away.
    auto issue = [&](int kb, int buf) {
        const int k0 = kb << 5;                    // BK*kb
        const float* a = pA + k0;
        const float* b = pB + (size_t)k0 * N;
        ASYNC_B128(ldsA[buf][0], a);
        ASYNC_B128(ldsA[buf][1], a + 16);
        ASYNC_B128(ldsB[buf][0], b);
        ASYNC_B128(ldsB[buf][1], b + b8N);
        ASYNC_B128(ldsB[buf][2], b + 2 * b8N);
        ASYNC_B128(ldsB[buf][3], b + 3 * b8N);
    };

    v8f acc00{}, acc01{}, acc10{}, acc11{};

    // f32 WMMA A(16x4): lanes 0-15 hold K={0,1}, lanes 16-31 K={2,3}
    const int mrow = lane & 15;
    const int kp   = (lane >> 4) << 1;

    auto compute = [&](int cur) {
        const float (*A)[BK + APAD] = As[cur];
        const float (*Bb)[BN + BPAD] = Bs[cur];
        #pragma unroll
        for (int ks = 0; ks < BK; ks += 4) {
            v2f a0, a1, b0, b1;
            a0.x = A[wm + mrow][ks + kp];
            a0.y = A[wm + mrow][ks + kp + 1];
            a1.x = A[wm + 16 + mrow][ks + kp];
            a1.y = A[wm + 16 + mrow][ks + kp + 1];
            b0.x = Bb[ks + kp][wn + mrow];
            b0.y = Bb[ks + kp + 1][wn + mrow];
            b1.x = Bb[ks + kp][wn + 16 + mrow];
            b1.y = Bb[ks + kp + 1][wn + 16 + mrow];
            acc00 = __builtin_amdgcn_wmma_f32_16x16x4_f32(false, a0, false, b0, (short)0, acc00, false, false);
            acc01 = __builtin_amdgcn_wmma_f32_16x16x4_f32(false, a0, false, b1, (short)0, acc01, false, false);
            acc10 = __builtin_amdgcn_wmma_f32_16x16x4_f32(false, a1, false, b0, (short)0, acc10, false, false);
            acc11 = __builtin_amdgcn_wmma_f32_16x16x4_f32(false, a1, false, b1, (short)0, acc11, false, false);
        }
    };

    const int nkb = K >> 5;          // even at every call site
    issue(0, 0);
    for (int kb = 0; kb < nkb; kb += 2) {
        issue(kb + 1, 1);
        asm volatile("s_wait_asynccnt 6" ::: "memory");   // stage kb ready
        __syncthreads();
        compute(0);
        __syncthreads();
        if (kb + 2 < nkb) {
            issue(kb + 2, 0);
            asm volatile("s_wait_asynccnt 6" ::: "memory"); // stage kb+1 ready
        } else {
            asm volatile("s_wait_asynccnt 0" ::: "memory");
        }
        __syncthreads();
        compute(1);
        __syncthreads();
    }

    // ---- epilogue: C/D layout = VGPR r: lanes0-15 M=r, lanes16-31 M=r+8 ----
    const int nl    = lane & 15;
    const int rhalf = (lane >> 4) << 3;
    #pragma unroll
    for (int mi2 = 0; mi2 < 2; ++mi2) {
        #pragma unroll
        for (int ni = 0; ni < 2; ++ni) {
            v8f acc = (mi2 == 0) ? (ni == 0 ? acc00 : acc01)
                                 : (ni == 0 ? acc10 : acc11);
            int n = n0 + wn + ni * 16 + nl;
            if (n >= N) continue;
            float bv = bias ? bias[n] : 0.f;
            #pragma unroll
            for (int r = 0; r < 8; ++r) {
                int m = m0 + wm + mi2 * 16 + r + rhalf;
                if (m < M) {
                    int row = m;
                    if (opad) { int b = m / T; row = m + 2 * b + 1; }
                    size_t idx = (size_t)row * N + n;
                    float val  = acc[r] + bv;
                    Y[idx] = accum ? (Y[idx] + val) : val;
                }
            }
        }
    }
}

// x (B,T,C) -> xp (B,T+2,C) with zero rows at t=0 and t=T+1
__global__ void pad_input(const float* __restrict__ x, float* __restrict__ xp,
                          int B_, int T, int C_)
{
    size_t Tp = (size_t)T + 2;
    size_t total = (size_t)B_ * Tp * C_;
    size_t i = (size_t)blockIdx.x * 256 + threadIdx.x;
    if (i >= total) return;
    int c = (int)(i % C_);
    size_t r = i / C_;
    int t = (int)(r % Tp);
    int b = (int)(r / Tp);
    xp[i] = (t == 0 || t == (int)Tp - 1) ? 0.f
                                         : x[((size_t)b * T + (t - 1)) * C_ + c];
}

// zero rows 0 and Tp-1 of each batch in a (B,Tp,N) buffer; grid = B*2*N/256
__global__ void zero_border(float* __restrict__ y, int Tp, int N)
{
    int i = blockIdx.x * 256 + threadIdx.x;
    int n = i % N;
    int r = i / N;            // 0..2B-1
    int b = r >> 1;
    int side = r & 1;
    y[((size_t)b * Tp + (side ? (size_t)(Tp - 1) : 0)) * N + n] = 0.f;
}

// w: (N, Cin, 3) -> wt: (3*Cin, N) row-major  (Cin = 1<<cshift)
__global__ void transpose_conv_w(const float* __restrict__ w, float* __restrict__ wt,
                                 int cshift, int N)
{
    int Cin = 1 << cshift;
    size_t total = (size_t)3 * Cin * N;
    size_t i = (size_t)blockIdx.x * 256 + threadIdx.x;
    if (i >= total) return;
    int    n   = (int)(i % N);
    size_t k   = i / N;
    int    tap = (int)(k >> cshift);
    int    ci  = (int)(k & (Cin - 1));
    wt[i] = w[((size_t)n * Cin + ci) * 3 + tap];
}

// w: (N, K) -> wt: (K, N)
__global__ void transpose_lin_w(const float* __restrict__ w, float* __restrict__ wt,
                                int K, int N)
{
    int i = blockIdx.x * 256 + threadIdx.x;
    if (i >= K * N) return;
    int n = i % N;
    int k = i / N;
    wt[i] = w[n * K + k];
}

// In-place ReLU + InstanceNorm over time: Y is (B,Tp,N) with `off` pad rows
// at the top; stats over the T interior rows per (b,n).
__global__ __launch_bounds__(256)
void relu_instnorm(float* __restrict__ Y, int T, int Tp, int off, int N)
{
    __shared__ float s1[4][64], s2[4][64], mstat[64], rstat[64];
    const int nl = threadIdx.x & 63;
    const int sl = threadIdx.x >> 6;
    const int n  = blockIdx.x * 64 + nl;
    const int b  = blockIdx.y;
    const size_t base = ((size_t)b * Tp + off) * N + n;

    float sum = 0.f, sq = 0.f;
    for (int t = sl; t < T; t += 4) {
        float v = fmaxf(Y[base + (size_t)t * N], 0.f);
        sum += v; sq += v * v;
    }
    s1[sl][nl] = sum; s2[sl][nl] = sq;
    __syncthreads();
    if (sl == 0) {
        float S = s1[0][nl] + s1[1][nl] + s1[2][nl] + s1[3][nl];
        float Q = s2[0][nl] + s2[1][nl] + s2[2][nl] + s2[3][nl];
        float m = S / (float)T;
        float var = Q / (float)T - m * m;
        mstat[nl] = m;
        rstat[nl] = rsqrtf(var + 1e-5f);
    }
    __syncthreads();
    float m = mstat[nl], r = rstat[nl];
    for (int t = sl; t < T; t += 4) {
        size_t i = base + (size_t)t * N;
        float v = fmaxf(Y[i], 0.f);
        Y[i] = (v - m) * r;
    }
}

// duration head: one wave per row, shuffle reduction; dur = floor(max(d,1))
__global__ __launch_bounds__(256)
void dur_linear(const float* __restrict__ h2, const float* __restrict__ wl,
                const float* __restrict__ bl, int* __restrict__ dur, int M)
{
    const int wave = threadIdx.x >> 5;
    const int lane = threadIdx.x & 31;
    const int m = blockIdx.x * 8 + wave;
    if (m >= M) return;
    float s = 0.f;
    for (int c = lane; c < 256; c += 32)
        s += h2[(size_t)m * 256 + c] * wl[c];
    #pragma unroll
    for (int off = 16; off; off >>= 1)
        s += __shfl_xor(s, off, 32);
    if (lane == 0) {
        float d = s + bl[0];
        dur[m] = (int)floorf(fmaxf(d, 1.0f));
    }
}

// inclusive cumsum of durations per batch (Hillis-Steele in LDS), block=512
__global__ void scan_dur(const int* __restrict__ dur, int* __restrict__ ends, int T)
{
    __shared__ int s[512];
    const int t = threadIdx.x;
    s[t] = dur[blockIdx.x * T + t];
    __syncthreads();
    for (int off = 1; off < 512; off <<= 1) {
        int v = (t >= off) ? s[t - off] : 0;
        __syncthreads();
        s[t] += v;
        __syncthreads();
    }
    ends[blockIdx.x * T + t] = s[t];
}

// expand: out[b,l,c] = att[b, searchsorted_right(ends_b, l) clamped, c]*(l<total)
// writes padded oex (for the conv predictors) and seeds d_out (base for +=).
__global__ void expand_kernel(const float* __restrict__ att, const int* __restrict__ ends,
                              float* __restrict__ oex, float* __restrict__ out,
                              int T, int L, int C)
{
    const int b = blockIdx.y, l = blockIdx.x, c = threadIdx.x;
    const int* e = ends + b * T;
    int lo = 0, hi = T;                     // first i with e[i] > l
    while (lo < hi) { int mid = (lo + hi) >> 1; if (e[mid] > l) hi = mid; else lo = mid + 1; }
    int idx = lo < T - 1 ? lo : T - 1;
    float mask = (l < e[T - 1]) ? 1.f : 0.f;
    float v = att[((size_t)(b * T + idx)) * C + c] * mask;
    oex[((size_t)b * (L + 2) + l + 1) * C + c] = v;       // padded interior
    out[((size_t)b * L + l) * C + c] = v;
}

extern "C" void kernel_launch(void* const* d_in, const int* in_sizes, int n_in,
                              void* d_out, int out_size, void* d_ws, size_t ws_size,
                              hipStream_t stream)
{
    const int B = 16, T = 512, C = 256, C4 = 1024;

    // pytree flatten order: att_out, then params sorted alphabetically:
    // dur, energy, last, pitch; within each: b1, b2, bl, w1, w2, wl.
    const float* att = (const float*)d_in[0];
    struct Pred { const float *b1, *b2, *bl, *w1, *w2, *wl; };
    auto grab = [&](int s) {
        return Pred{ (const float*)d_in[s],     (const float*)d_in[s + 1],
                     (const float*)d_in[s + 2], (const float*)d_in[s + 3],
                     (const float*)d_in[s + 4], (const float*)d_in[s + 5] };
    };
    Pred Pd = grab(1), Pe = grab(7), Pl = grab(13), Pp = grab(19);

    const int  L  = out_size / (B * C);
    const int  M1 = B * T;
    const long M2 = (long)B * L;
    const long Mh = (M2 > M1) ? M2 : M1;

    float* ws = (float*)d_ws;
    size_t o = 0;
    float* attp = ws + o; o += (size_t)B * (T + 2) * C;
    float* wt1  = ws + o; o += (size_t)3 * C  * C4;
    float* wt2  = ws + o; o += (size_t)3 * C4 * C;
    float* wlt  = ws + o; o += (size_t)C * C;
    float* h1p  = ws + o; o += (size_t)B * (L + 2) * C4;
    float* h2   = ws + o; o += (size_t)Mh * C;
    float* oexp = ws + o; o += (size_t)B * (L + 2) * C;
    int* durv   = (int*)(ws + o); o += M1;
    int* endsv  = (int*)(ws + o); o += M1;
    (void)ws_size; (void)in_sizes; (void)n_in;

    float* out = (float*)d_out;

    auto gemm = [&](const float* X, const float* Wt, const float* bias, float* Y,
                    long M, int Tn, int Tpn, int cshift, int N, int K,
                    int conv, int opad, int accum) {
        dim3 g((unsigned)((N + BN - 1) / BN), (unsigned)((M + BM - 1) / BM));
        wmma_gemm<<<g, 256, 0, stream>>>(X, Wt, bias, Y, (int)M, Tn, Tpn,
                                         cshift, N, K, conv, opad, accum);
    };

    // ---------------- duration predictor on att_out ----------------
    pad_input<<<((size_t)B * (T + 2) * C + 255) / 256, 256, 0, stream>>>(att, attp, B, T, C);
    transpose_conv_w<<<(3 * C * C4 + 255) / 256, 256, 0, stream>>>(Pd.w1, wt1, 8, C4);
    transpose_conv_w<<<(3 * C4 * C + 255) / 256, 256, 0, stream>>>(Pd.w2, wt2, 10, C);
    zero_border<<<(B * 2 * C4) / 256, 256, 0, stream>>>(h1p, T + 2, C4);
    gemm(attp, wt1, Pd.b1, h1p, M1, T, T + 2, 8, C4, 3 * C, 1, 1, 0);
    relu_instnorm<<<dim3(C4 / 64, B), 256, 0, stream>>>(h1p, T, T + 2, 1, C4);
    gemm(h1p, wt2, Pd.b2, h2, M1, T, T + 2, 10, C, 3 * C4, 1, 0, 0);
    relu_instnorm<<<dim3(C / 64, B), 256, 0, stream>>>(h2, T, T, 0, C);
    dur_linear<<<M1 / 8, 256, 0, stream>>>(h2, Pd.wl, Pd.bl, durv, M1);
    scan_dur<<<B, T, 0, stream>>>(durv, endsv, T);
    zero_border<<<(B * 2 * C) / 256, 256, 0, stream>>>(oexp, L + 2, C);
    expand_kernel<<<dim3(L, B), 256, 0, stream>>>(att, endsv, oexp, out, T, L, C);
    zero_border<<<(B * 2 * C4) / 256, 256, 0, stream>>>(h1p, L + 2, C4);

    // ------------- pitch / energy / last on expanded tensor -------------
    const Pred preds[3] = { Pp, Pe, Pl };
    for (int i = 0; i < 3; ++i) {
        const Pred& p = preds[i];
        transpose_conv_w<<<(3 * C * C4 + 255) / 256, 256, 0, stream>>>(p.w1, wt1, 8, C4);
        transpose_conv_w<<<(3 * C4 * C + 255) / 256, 256, 0, stream>>>(p.w2, wt2, 10, C);
        gemm(oexp, wt1, p.b1, h1p, M2, L, L + 2, 8, C4, 3 * C, 1, 1, 0);
        relu_instnorm<<<dim3(C4 / 64, B), 256, 0, stream>>>(h1p, L, L + 2, 1, C4);
        gemm(h1p, wt2, p.b2, h2, M2, L, L + 2, 10, C, 3 * C4, 1, 0, 0);
        relu_instnorm<<<dim3(C / 64, B), 256, 0, stream>>>(h2, L, L, 0, C);
        transpose_lin_w<<<(C * C + 255) / 256, 256, 0, stream>>>(p.wl, wlt, C, C);
        gemm(h2, wlt, p.bl, out, M2, L, L, 0, C, C, 0, 0, 1);  // out += h2@wl.T + bl
    }
}